// OneClassLoss_21251498181004
// MI455X (gfx1250) — compile-verified
//
#include <hip/hip_runtime.h>
#include <math.h>

// ---------------------------------------------------------------------------
// OneClassLoss on MI455X (gfx1250, wave32).
//   dot   = Xf @ Xf^T            -> V_WMMA_F32_16X16X4_F32, K = 65536
//   ce    = log-softmax over gathered distances (VALU)
//   psd   = 2D DFT as dense matmuls with symmetric DFT matrix E (WMMA f32),
//           x_b panels staged into LDS by the Tensor Data Mover (TDM),
//           double-buffered on TENSORcnt.
// x (52 MB) fits in the 192 MB L2 -> compute bound; f32 WMMA for precision.
// ---------------------------------------------------------------------------

typedef float        v2f __attribute__((ext_vector_type(2)));
typedef float        v8f __attribute__((ext_vector_type(8)));
typedef unsigned int v4u __attribute__((ext_vector_type(4)));
typedef int          v4i __attribute__((ext_vector_type(4)));
typedef int          v8i __attribute__((ext_vector_type(8)));

#define NB     200      // 2*B samples
#define FEAT   65536    // H*W
#define HH     256      // H = W
#define LDPAD  208      // padded leading dim for the 200x200 gram matrix
#define NROWS  600      // mask rows
#define NCOLS  197      // logits per row
#define EPSC   1e-8f

__device__ __forceinline__ v8f wmma4_f32(v2f a, v2f b, v8f c) {
    // 8 args: (neg_a, A, neg_b, B, c_mod, C, reuse_a, reuse_b)
    return __builtin_amdgcn_wmma_f32_16x16x4_f32(
        false, a, false, b, (short)0, c, false, false);
}

// ---------------------------------------------------------------------------
// TDM: DMA a 16-row x 256-col f32 panel (16 KB) from global into LDS.
// D# per CDNA5 ISA ch.8: the panel is described as its own 2-D tensor
// (tensor == tile -> no OOB path). Tracked with TENSORcnt.
// This toolchain exposes the 6-arg (clang-23 / therock-10.0) builtin:
//   (v4u group0, v8i group1, v4i group2, v4i group3, v8i group4, i32 cpol)
// Groups 2..4 are unused for a plain 2-D transfer -> zeros.
// ---------------------------------------------------------------------------
__device__ __forceinline__ void tdm_load_panel_16x256(const float* gsrc,
                                                      unsigned ldsAddr) {
    unsigned long long ga = (unsigned long long)(size_t)gsrc;
    v4u g0;
    g0.x = 1u;                                        // count=1 (valid user D#)
    g0.y = ldsAddr;                                   // lds_addr [63:32]
    g0.z = (unsigned)(ga & 0xFFFFFFFFu);              // global_addr [95:64]
    g0.w = (unsigned)((ga >> 32) & 0x1FFFFFFu)        // global_addr [120:96]
         | (2u << 30);                                // type=2 ("image") [127:126]
    v8i g1;
    g1[0] = (int)(2u << 16);      // workgroup_mask=0, data_size=2 (4 bytes)
    g1[1] = (int)(256u << 16);    // tensor_dim0 = 256      (bits 79:48)
    g1[2] = (int)(16u << 16);     // tensor_dim1 = 16       (bits 111:80)
    g1[3] = (int)(256u << 16);    // tile_dim0   = 256      (bits 127:112)
    g1[4] = 16;                   // tile_dim1   = 16       (bits 143:128)
    g1[5] = 256;                  // tensor_dim0_stride=256 (bits 207:160)
    g1[6] = 0;
    g1[7] = 0;
    v4i z4 = {0, 0, 0, 0};        // groups 2/3 unused (2-D tensor)
    v8i z8 = {0, 0, 0, 0, 0, 0, 0, 0};
    __builtin_amdgcn_tensor_load_to_lds(g0, g1, z4, z4, z8, 0);
}

// ---------------------------------------------------------------------------
// K0: twiddle tables E = cos(th) + i*(-sin(th)), th = 2*pi*(u*h mod 256)/256;
//     zero the avgpsd and CE accumulators (must happen every launch: the
//     harness graph-replays without re-poisoning).
// ---------------------------------------------------------------------------
__global__ void ocl_init_kernel(float* __restrict__ trigC, float* __restrict__ trigS,
                                float* __restrict__ avgpsd, float* __restrict__ ceSum) {
    int idx = blockIdx.x * blockDim.x + threadIdx.x;
    if (idx < FEAT) {
        int u = idx >> 8, h = idx & 255;
        int ph = (u * h) & 255;
        float ang = -6.28318530717958647692f * (float)ph * (1.0f / 256.0f);
        trigC[idx] = cosf(ang);   // Re(E)
        trigS[idx] = sinf(ang);   // Im(E) = -sin(th)
        avgpsd[idx] = 0.0f;
    }
    if (idx == 0) ceSum[0] = 0.0f;
}

// ---------------------------------------------------------------------------
// K1: gram matrix dot = Xf @ Xf^T via WMMA f32 16x16x4.
// Grid (13,13) 16x16 tiles (padded to 208); 8 waves each own a K-slice of
// 8192 and accumulate privately, then reduce through LDS.
// A fragment: lane holds row m=lane%16, K pair at 2*(lane/16)+{0,1}.
// B fragment: lane holds col n=lane%16 of X^T -> row (j0+n) of X, same K pair.
// Both are contiguous float2 (b64) loads; everything hits L2.
// ---------------------------------------------------------------------------
__global__ void ocl_gram_kernel(const float* __restrict__ X, float* __restrict__ dotm) {
    __shared__ float part[8][256];
    const int lane = threadIdx.x & 31;
    const int wave = threadIdx.x >> 5;
    const int i0 = blockIdx.x * 16;
    const int j0 = blockIdx.y * 16;

    int mrow = i0 + (lane & 15); if (mrow > NB - 1) mrow = NB - 1; // clamp pad rows
    int nrow = j0 + (lane & 15); if (nrow > NB - 1) nrow = NB - 1;
    const int kh = (lane >> 4) << 1;           // 0 or 2
    const float* Aptr = X + (size_t)mrow * FEAT + kh;
    const float* Bptr = X + (size_t)nrow * FEAT + kh;

    const int kbeg = wave * (FEAT / 8);
    const int kend = kbeg + (FEAT / 8);
    __builtin_prefetch(Aptr + kbeg, 0, 0);
    __builtin_prefetch(Bptr + kbeg, 0, 0);

    v8f acc = {0.f, 0.f, 0.f, 0.f, 0.f, 0.f, 0.f, 0.f};
    for (int k = kbeg; k < kend; k += 4) {
        v2f a = *(const v2f*)(Aptr + k);
        v2f b = *(const v2f*)(Bptr + k);
        acc = wmma4_f32(a, b, acc);
    }
    #pragma unroll
    for (int e = 0; e < 8; ++e) part[wave][lane * 8 + e] = acc[e];
    __syncthreads();

    // cross-wave reduce: thread t owns C slot (lane=t>>3, elem=t&7)
    const int l2 = threadIdx.x >> 3;
    const int e2 = threadIdx.x & 7;
    float s = 0.f;
    #pragma unroll
    for (int w = 0; w < 8; ++w) s += part[w][l2 * 8 + e2];
    const int m = e2 + ((l2 >> 4) << 3);   // C layout: VGPR e -> M = e + 8*(lane/16)
    const int n = l2 & 15;
    dotm[(i0 + m) * LDPAD + (j0 + n)] = s;
}

// ---------------------------------------------------------------------------
// K2: cross-entropy. One block per mask row; distances rebuilt from dot.
// ---------------------------------------------------------------------------
__global__ void ocl_ce_kernel(const float* __restrict__ dotm, const int* __restrict__ masks,
                              const int* __restrict__ labels, float* __restrict__ ceSum) {
    __shared__ float slog[256];
    __shared__ float red[256];
    const int r = blockIdx.x;
    const int i = r / 3;
    const int t = threadIdx.x;

    const float sqi = dotm[i * LDPAD + i];
    float lg = -INFINITY;
    if (t < NCOLS) {
        int j = masks[r * NCOLS + t];
        float sqj = dotm[j * LDPAD + j];
        float dd = sqi + sqj - 2.0f * dotm[i * LDPAD + j];
        dd = fmaxf(dd, 0.0f);
        float z = (dd == 0.0f) ? 1.0f : 0.0f;
        lg = sqrtf(dd + z * EPSC) * (1.0f - z);
    }
    slog[t] = lg;
    red[t] = lg;
    __syncthreads();
    for (int s = 128; s > 0; s >>= 1) {
        if (t < s) red[t] = fmaxf(red[t], red[t + s]);
        __syncthreads();
    }
    const float mx = red[0];
    __syncthreads();
    red[t] = (t < NCOLS) ? expf(slog[t] - mx) : 0.0f;
    __syncthreads();
    for (int s = 128; s > 0; s >>= 1) {
        if (t < s) red[t] += red[t + s];
        __syncthreads();
    }
    if (t == 0) {
        float lse = mx + logf(red[0]);
        float ll = slog[labels[r]];
        atomicAdd(ceSum, lse - ll);
    }
}

// ---------------------------------------------------------------------------
// K3: PSD accumulation. Block = (u-band of 16 rows, image group).
// Stage 1: U(16x256) = E[u-band,:] @ x_b. x_b is consumed as 16 row-panels
//          of 16x256 f32, DMA'd into LDS by the TDM (double-buffered,
//          TENSORcnt-tracked, issued by wave 0 only); B fragments then come
//          from LDS. U is written to LDS (row stride 258 to spread banks).
// Stage 2: dft(16x16 tiles) = U @ E, 4 WMMAs per K-step:
//          re += Ur*Er - Ui*Ei ; im += Ur*Ei + Ui*Er
// |dft|^2 accumulated in registers over ~13 images, one atomicAdd/elem at end.
// ---------------------------------------------------------------------------
__global__ void ocl_psd_kernel(const float* __restrict__ X, const float* __restrict__ Cre,
                               const float* __restrict__ Cim, float* __restrict__ avgpsd) {
    __shared__ float Ure[16][258];
    __shared__ float Uim[16][258];
    __shared__ float xpanel[2][16 * HH];        // TDM destination, 2 x 16 KB

    const int lane = threadIdx.x & 31;
    const int wave = threadIdx.x >> 5;
    const int u0 = blockIdx.x * 16;
    const int mIdx = lane & 15;
    const int kh = (lane >> 4) << 1;
    const int n = lane & 15;
    const int n0a = wave * 32;                  // this wave's two N-tiles
    const int n0b = wave * 32 + 16;

    const unsigned ldsP0 = (unsigned)(size_t)&xpanel[0][0];
    const unsigned ldsP1 = (unsigned)(size_t)&xpanel[1][0];

    const float* Are = Cre + (u0 + mIdx) * HH + kh;   // stage-1 A rows (E band)
    const float* Aim = Cim + (u0 + mIdx) * HH + kh;

    v8f psdAcc0 = {0.f, 0.f, 0.f, 0.f, 0.f, 0.f, 0.f, 0.f};
    v8f psdAcc1 = psdAcc0;

    for (int b = blockIdx.y; b < NB; b += gridDim.y) {
        const float* xb = X + (size_t)b * FEAT;
        __syncthreads();                       // LDS reuse fence across images

        // ---- stage 1: U = E @ x_b, x_b panels via TDM double-buffer
        v8f accR0 = {0.f, 0.f, 0.f, 0.f, 0.f, 0.f, 0.f, 0.f};
        v8f accI0 = accR0, accR1 = accR0, accI1 = accR0;

        if (threadIdx.x < 32) {                // wave 0 primes panel 0
            tdm_load_panel_16x256(xb, ldsP0);
            __builtin_amdgcn_s_wait_tensorcnt(0);
        }
        __syncthreads();

        for (int p = 0; p < 16; ++p) {
            if (p < 15 && threadIdx.x < 32)    // kick next panel while computing
                tdm_load_panel_16x256(xb + (size_t)(p + 1) * 16 * HH,
                                      (p & 1) ? ldsP0 : ldsP1);
            const float* pan = &xpanel[p & 1][0];
            #pragma unroll
            for (int ks = 0; ks < 4; ++ks) {
                const int kb = p * 16 + ks * 4;
                const int lrow = ks * 4 + kh;          // row inside panel
                v2f ar = *(const v2f*)(Are + kb);
                v2f ai = *(const v2f*)(Aim + kb);
                v2f b0, b1;
                b0.x = pan[lrow * HH + n0a + n];
                b0.y = pan[(lrow + 1) * HH + n0a + n];
                b1.x = pan[lrow * HH + n0b + n];
                b1.y = pan[(lrow + 1) * HH + n0b + n];
                accR0 = wmma4_f32(ar, b0, accR0);
                accI0 = wmma4_f32(ai, b0, accI0);
                accR1 = wmma4_f32(ar, b1, accR1);
                accI1 = wmma4_f32(ai, b1, accI1);
            }
            if (threadIdx.x < 32)              // next panel landed?
                __builtin_amdgcn_s_wait_tensorcnt(0);
            __syncthreads();
        }

        #pragma unroll
        for (int e = 0; e < 8; ++e) {
            const int m = e + ((lane >> 4) << 3);
            Ure[m][n0a + n] = accR0[e];
            Uim[m][n0a + n] = accI0[e];
            Ure[m][n0b + n] = accR1[e];
            Uim[m][n0b + n] = accI1[e];
        }
        __syncthreads();

        // ---- stage 2: this wave computes 2 v-tiles of dft, squares into psdAcc
        #pragma unroll
        for (int q = 0; q < 2; ++q) {
            const int v0 = (wave * 2 + q) * 16;
            v8f accR = {0.f, 0.f, 0.f, 0.f, 0.f, 0.f, 0.f, 0.f};
            v8f accI = accR;
            for (int kb = 0; kb < HH; kb += 4) {
                const int kk = kb + kh;
                v2f ur = *(const v2f*)(&Ure[mIdx][kk]);
                v2f ui = *(const v2f*)(&Uim[mIdx][kk]);
                v2f br, bi, bin;
                br.x = Cre[kk * HH + v0 + n];
                br.y = Cre[(kk + 1) * HH + v0 + n];
                bi.x = Cim[kk * HH + v0 + n];
                bi.y = Cim[(kk + 1) * HH + v0 + n];
                bin.x = -bi.x; bin.y = -bi.y;
                accR = wmma4_f32(ur, br,  accR);
                accR = wmma4_f32(ui, bin, accR);   // - Ui*Ei
                accI = wmma4_f32(ur, bi,  accI);
                accI = wmma4_f32(ui, br,  accI);
            }
            v8f p2 = accR * accR + accI * accI;
            if (q == 0) psdAcc0 += p2; else psdAcc1 += p2;
        }
    }

    // flush |dft|^2 sums (16 image-groups contend -> atomics)
    #pragma unroll
    for (int q = 0; q < 2; ++q) {
        const int v0 = (wave * 2 + q) * 16;
        #pragma unroll
        for (int e = 0; e < 8; ++e) {
            const int m = e + ((lane >> 4) << 3);
            float v = (q == 0) ? psdAcc0[e] : psdAcc1[e];
            atomicAdd(&avgpsd[(u0 + m) * HH + v0 + n], v);
        }
    }
}

// ---------------------------------------------------------------------------
// K4: final reduction. Batch-mean scaling of avgpsd cancels exactly in
// mean(log(a)) - log(mean(a)), so the raw sum is used directly.
// ---------------------------------------------------------------------------
__global__ void ocl_final_kernel(const float* __restrict__ avgpsd,
                                 const float* __restrict__ ceSum,
                                 float* __restrict__ out) {
    __shared__ float sL[256];
    __shared__ float sV[256];
    const int t = threadIdx.x;
    float aL = 0.f, aV = 0.f;
    for (int i = t; i < FEAT; i += 256) {
        float v = avgpsd[i];
        aL += logf(v);
        aV += v;
    }
    sL[t] = aL; sV[t] = aV;
    __syncthreads();
    for (int s = 128; s > 0; s >>= 1) {
        if (t < s) { sL[t] += sL[t + s]; sV[t] += sV[t + s]; }
        __syncthreads();
    }
    if (t == 0) {
        float psd = sL[0] * (1.0f / 65536.0f) - logf(sV[0] * (1.0f / 65536.0f));
        out[0] = ceSum[0] * (1.0f / 600.0f) - 0.1f * psd;
    }
}

// ---------------------------------------------------------------------------
extern "C" void kernel_launch(void* const* d_in, const int* in_sizes, int n_in,
                              void* d_out, int out_size, void* d_ws, size_t ws_size,
                              hipStream_t stream) {
    const float* X      = (const float*)d_in[0];   // (200,1,256,256) f32
    const int*   masks  = (const int*)d_in[1];     // (600,197) i32
    const int*   labels = (const int*)d_in[2];     // (600,) i32
    float* out = (float*)d_out;

    float* wsf    = (float*)d_ws;
    float* trigC  = wsf;                           // 65536
    float* trigS  = wsf + 65536;                   // 65536
    float* avgpsd = wsf + 131072;                  // 65536
    float* dotm   = wsf + 196608;                  // 208*208 = 43264
    float* ceSum  = wsf + 196608 + LDPAD * LDPAD;  // 1

    ocl_init_kernel<<<FEAT / 256, 256, 0, stream>>>(trigC, trigS, avgpsd, ceSum);
    ocl_gram_kernel<<<dim3(13, 13), 256, 0, stream>>>(X, dotm);
    ocl_ce_kernel<<<NROWS, 256, 0, stream>>>(dotm, masks, labels, ceSum);
    ocl_psd_kernel<<<dim3(16, 16), 256, 0, stream>>>(X, trigC, trigS, avgpsd);
    ocl_final_kernel<<<1, 256, 0, stream>>>(avgpsd, ceSum, out);
}